// HGTConv_360777253416
// MI455X (gfx1250) — compile-verified
//
#include <hip/hip_runtime.h>
#include <math.h>

#define N_V 100000
#define M_E 50000
#define P_P 1600000
#define C   128
#define NEG_SLOPE 0.2f

typedef __attribute__((ext_vector_type(2))) float v2f;
typedef __attribute__((ext_vector_type(8))) float v8f;

// ---------------- WMMA GEMM: H = X @ W^T + b  (fp32, exact) ----------------
// grid.x = N/16 row tiles, block = (32, 8): wave y handles column tile y (8 tiles of 16 = 128 cols).
__global__ __launch_bounds__(256)
void k_gemm_h(const float* __restrict__ X, const float* __restrict__ W,
              const float* __restrict__ bias, float* __restrict__ H) {
    const int lane = threadIdx.x;           // 0..31
    const int ct   = threadIdx.y;           // column tile 0..7
    const int rt   = blockIdx.x;            // row tile
    const int mr   = lane & 15;
    const int hi   = lane >> 4;             // 0 or 1 (lane half)
    const int row  = rt * 16 + mr;          // A-matrix row (M)
    const int ncol = ct * 16 + mr;          // B-matrix col (N) == output channel
    const int koff = hi * 2;                // K offset of this lane half

    // A 16x4 fp32 layout: lanes 0-15 M=lane,K={k,k+1}; lanes 16-31 M=lane-16,K={k+2,k+3}
    // B 4x16 fp32 layout (B[k][n] = W[n][k]): same K striping, N = lane&15
    const v2f* __restrict__ xp = (const v2f*)(X + (size_t)row  * C + koff);
    const v2f* __restrict__ wp = (const v2f*)(W + (size_t)ncol * C + koff);

    v8f acc = {};
#pragma unroll
    for (int t = 0; t < C / 4; ++t) {
        v2f a = xp[2 * t];
        v2f b = wp[2 * t];
        acc = __builtin_amdgcn_wmma_f32_16x16x4_f32(
            /*neg_a=*/false, a, /*neg_b=*/false, b,
            /*c_mod=*/(short)0, acc, /*reuse_a=*/false, /*reuse_b=*/false);
    }

    const float bb = bias[ncol];
    // C/D layout: VGPR v -> M = v + hi*8, N = lane&15
#pragma unroll
    for (int v = 0; v < 8; ++v) {
        const int orow = rt * 16 + v + hi * 8;
        H[(size_t)orow * C + ncol] = acc[v] + bb;
    }
}

// ---------------- s = H.w_src, d = H.w_dst (one wave per row) ----------------
__global__ __launch_bounds__(256)
void k_sd(const float* __restrict__ H, const float* __restrict__ wsrc,
          const float* __restrict__ wdst, float* __restrict__ s, float* __restrict__ d) {
    const int lane = threadIdx.x & 31;
    const int row  = (blockIdx.x * blockDim.x + threadIdx.x) >> 5;
    if (row >= N_V) return;
    const float4 h = *(const float4*)(H + (size_t)row * C + lane * 4);
    const float4 a = *(const float4*)(wsrc + lane * 4);
    const float4 b = *(const float4*)(wdst + lane * 4);
    float ps = h.x * a.x + h.y * a.y + h.z * a.z + h.w * a.w;
    float pd = h.x * b.x + h.y * b.y + h.z * b.z + h.w * b.w;
#pragma unroll
    for (int off = 16; off > 0; off >>= 1) {
        ps += __shfl_down(ps, off, 32);
        pd += __shfl_down(pd, off, 32);
    }
    if (lane == 0) { s[row] = ps; d[row] = pd; }
}

// ---------------- degrees ----------------
__global__ __launch_bounds__(256)
void k_deg(const int* __restrict__ pv, const int* __restrict__ pe,
           float* __restrict__ dvc, float* __restrict__ dec) {
    const int p = blockIdx.x * blockDim.x + threadIdx.x;
    if (p >= P_P) return;
    atomicAdd(dvc + pv[p], 1.0f);
    atomicAdd(dec + pe[p], 1.0f);
}

__global__ __launch_bounds__(256)
void k_prep_v(const float* __restrict__ dvc, float* __restrict__ dvis, float* __restrict__ idv) {
    const int n = blockIdx.x * blockDim.x + threadIdx.x;
    if (n >= N_V) return;
    const float dv = fmaxf(dvc[n], 1.0f);
    dvis[n] = rsqrtf(dv);
    idv[n]  = 1.0f / dv;
}

__global__ __launch_bounds__(256)
void k_prep_e(const float* __restrict__ dec, float* __restrict__ ide) {
    const int m = blockIdx.x * blockDim.x + threadIdx.x;
    if (m >= M_E) return;
    ide[m] = 1.0f / fmaxf(dec[m], 1.0f);
}

// ---------------- attention scores + segment max ----------------
__global__ __launch_bounds__(256)
void k_score(const int* __restrict__ srcv, const int* __restrict__ pv,
             const float* __restrict__ s, const float* __restrict__ d,
             float* __restrict__ score, int* __restrict__ segmax) {
    const int p = blockIdx.x * blockDim.x + threadIdx.x;
    if (p >= P_P) return;
    float x = s[srcv[p]] + d[pv[p]];
    x = (x >= 0.0f) ? x : NEG_SLOPE * x;          // leaky_relu
    x = fminf(fmaxf(x, 0.001f), 5.0f);            // clip -> strictly positive
    score[p] = x;
    atomicMax(segmax + pv[p], __float_as_int(x)); // int-max == float-max for x>0
}

__global__ __launch_bounds__(256)
void k_exp(const int* __restrict__ pv, const int* __restrict__ segmax,
           float* __restrict__ score, float* __restrict__ ssum) {
    const int p = blockIdx.x * blockDim.x + threadIdx.x;
    if (p >= P_P) return;
    const float m = __int_as_float(segmax[pv[p]]);
    const float e = expf(score[p] - m);
    score[p] = e;                                  // in place: score -> exp
    atomicAdd(ssum + pv[p], e);
}

__global__ __launch_bounds__(256)
void k_alpha(const int* __restrict__ pv, const float* __restrict__ ssum,
             float* __restrict__ score) {
    const int p = blockIdx.x * blockDim.x + threadIdx.x;
    if (p >= P_P) return;
    score[p] = score[p] / ssum[pv[p]];             // in place: exp -> alpha
}

// ---------------- generic pair scatter: dst[si] += w * src[gi] -------------
// one wave per pair; float4 per lane covers the 128-float row.
__global__ __launch_bounds__(256)
void k_scatter(const float* __restrict__ src, float* __restrict__ dst,
               const int* __restrict__ gi, const int* __restrict__ si,
               const float* __restrict__ rowscale,   // nullable, indexed by gi
               const float* __restrict__ pw) {       // nullable, per-pair
    const int  lane = threadIdx.x & 31;
    const long wid  = (long)blockIdx.x * (blockDim.x >> 5) + (threadIdx.x >> 5);
    if (wid >= P_P) return;
    const int g = gi[wid];
    const int o = si[wid];
    float w = 1.0f;
    if (rowscale) w *= rowscale[g];
    if (pw)       w *= pw[wid];
    const float4 v = *(const float4*)(src + (size_t)g * C + lane * 4);
    float* out = dst + (size_t)o * C + lane * 4;
    atomicAdd(out + 0, v.x * w);
    atomicAdd(out + 1, v.y * w);
    atomicAdd(out + 2, v.z * w);
    atomicAdd(out + 3, v.w * w);
}

// ---------------- finalize: out = elu(out * inv_dv) in place ---------------
__global__ __launch_bounds__(128)
void k_final(float* __restrict__ out, const float* __restrict__ idv) {
    const int n = blockIdx.x;
    const int k = threadIdx.x;
    const size_t i = (size_t)n * C + k;
    const float x = out[i] * idv[n];
    out[i] = (x > 0.0f) ? x : (expm1f(x));
}

extern "C" void kernel_launch(void* const* d_in, const int* in_sizes, int n_in,
                              void* d_out, int out_size, void* d_ws, size_t ws_size,
                              hipStream_t stream) {
    const float* X    = (const float*)d_in[0];
    const int*   pv   = (const int*)  d_in[1];   // pair_v
    const int*   pe   = (const int*)  d_in[2];   // pair_e
    const int*   sv   = (const int*)  d_in[3];   // src_v
    const float* W    = (const float*)d_in[4];   // W_theta [C_OUT, C_IN]
    const float* bth  = (const float*)d_in[5];
    const float* wsrc = (const float*)d_in[6];
    const float* wdst = (const float*)d_in[7];
    float* out = (float*)d_out;                  // N_V x C

    // ---- workspace carve-up (~86 MB) ----
    float* ws   = (float*)d_ws;
    float* H    = ws; ws += (size_t)N_V * C;     // H, later reused as Xv1
    float* bufM = ws; ws += (size_t)M_E * C;     // edge-feature accumulator
    float* s    = ws; ws += N_V;
    float* d    = ws; ws += N_V;
    float* dvc  = ws; ws += N_V;                 // vertex degree counts
    float* dvis = ws; ws += N_V;                 // dv^-1/2
    float* idv  = ws; ws += N_V;                 // 1/dv
    float* dec  = ws; ws += M_E;                 // edge degree counts
    float* ide  = ws; ws += M_E;                 // 1/de
    float* segm = ws; ws += N_V;                 // segment max (int bits)
    float* ssum = ws; ws += N_V;                 // segment exp-sum
    float* sc   = ws; ws += P_P;                 // score -> exp -> alpha

    const dim3 b256(256);
    const dim3 gP((P_P + 255) / 256);
    const dim3 gN((N_V + 255) / 256);
    const dim3 gM((M_E + 255) / 256);
    const dim3 gWaveP(P_P / 8);                  // one wave per pair, 8 waves/block
    const dim3 gWaveN((N_V * 32 + 255) / 256);   // one wave per row

    // ---- degrees ----
    hipMemsetAsync(dvc,  0, sizeof(float) * N_V, stream);
    hipMemsetAsync(dec,  0, sizeof(float) * M_E, stream);
    hipMemsetAsync(segm, 0, sizeof(float) * N_V, stream);
    hipMemsetAsync(ssum, 0, sizeof(float) * N_V, stream);
    k_deg<<<gP, b256, 0, stream>>>(pv, pe, dvc, dec);
    k_prep_v<<<gN, b256, 0, stream>>>(dvc, dvis, idv);
    k_prep_e<<<gM, b256, 0, stream>>>(dec, ide);

    // ---- projection + attention ----
    k_gemm_h<<<dim3(N_V / 16), dim3(32, 8), 0, stream>>>(X, W, bth, H);
    k_sd<<<gWaveN, b256, 0, stream>>>(H, wsrc, wdst, s, d);
    k_score<<<gP, b256, 0, stream>>>(sv, pv, s, d, sc, (int*)segm);
    k_exp<<<gP, b256, 0, stream>>>(pv, (int*)segm, sc, ssum);
    k_alpha<<<gP, b256, 0, stream>>>(pv, ssum, sc);

    // ---- pass A: Xe  = segsum((H*dvis)[pair_v] -> pair_e)      (de fold downstream)
    hipMemsetAsync(bufM, 0, sizeof(float) * (size_t)M_E * C, stream);
    k_scatter<<<gWaveP, b256, 0, stream>>>(H, bufM, pv, pe, dvis, nullptr);
    // ---- pass B: Hs  = segsum((Xe*ide)[pair_e] -> pair_v)      (dvis fold downstream)
    hipMemsetAsync(out, 0, sizeof(float) * (size_t)N_V * C, stream);
    k_scatter<<<gWaveP, b256, 0, stream>>>(bufM, out, pe, pv, ide, nullptr);
    // ---- pass C: Xe1 = segsum((Hs*dvis)[pair_v] -> pair_e)
    hipMemsetAsync(bufM, 0, sizeof(float) * (size_t)M_E * C, stream);
    k_scatter<<<gWaveP, b256, 0, stream>>>(out, bufM, pv, pe, dvis, nullptr);
    // ---- pass D: Xv1 = segsum(alpha * (Xe1*ide)[pair_e] -> pair_v)   (into H's buffer)
    hipMemsetAsync(H, 0, sizeof(float) * (size_t)N_V * C, stream);
    k_scatter<<<gWaveP, b256, 0, stream>>>(bufM, H, pe, pv, ide, sc);
    // ---- pass E: Xe2 = segsum(Xv1[pair_v] -> pair_e)
    hipMemsetAsync(bufM, 0, sizeof(float) * (size_t)M_E * C, stream);
    k_scatter<<<gWaveP, b256, 0, stream>>>(H, bufM, pv, pe, nullptr, nullptr);
    // ---- pass F: Xv2 = segsum((Xe2*ide)[pair_e] -> pair_v)
    hipMemsetAsync(out, 0, sizeof(float) * (size_t)N_V * C, stream);
    k_scatter<<<gWaveP, b256, 0, stream>>>(bufM, out, pe, pv, ide, nullptr);

    // ---- finalize: out = elu(Xv2 / dv) ----
    k_final<<<dim3(N_V), dim3(C), 0, stream>>>(out, idv);
}